// MultiBoxLoss_22230750724470
// MI455X (gfx1250) — compile-verified
//
#include <hip/hip_runtime.h>
#include <hip/hip_bf16.h>
#include <stdint.h>

#define NUM_B 64
#define NUM_P 8732
#define NUM_O 50
#define NUM_C 81

#if defined(__has_builtin)
#  if __has_builtin(__builtin_amdgcn_global_load_async_to_lds_b128)
#    define HAVE_ASYNC_LDS 1
#  else
#    define HAVE_ASYNC_LDS 0
#  endif
#  if __has_builtin(__builtin_amdgcn_s_wait_asynccnt)
#    define HAVE_WAIT_ASYNC 1
#  else
#    define HAVE_WAIT_ASYNC 0
#  endif
#else
#  define HAVE_ASYNC_LDS 0
#  define HAVE_WAIT_ASYNC 0
#endif

// Pointer types matching the async-to-LDS builtin signature discovered from
// the compiler diagnostic: param0 is a global (AS1) pointer to a 16-byte
// int vector; param1 is the LDS (AS3) side.
typedef int v4i_t __attribute__((ext_vector_type(4)));
typedef __attribute__((address_space(1))) v4i_t* gptr_v4i;
typedef __attribute__((address_space(3))) v4i_t* lptr_v4i;

__device__ __forceinline__ void wait_asynccnt0() {
#if HAVE_WAIT_ASYNC
    __builtin_amdgcn_s_wait_asynccnt(0);
#else
    asm volatile("s_wait_asynccnt 0x0" ::: "memory");
#endif
}

// ---------------- Kernel 0: zero the global accumulators ----------------
__global__ void k_init(float* accums) {
    if (threadIdx.x < 2) accums[threadIdx.x] = 0.0f;
}

// ---------------- Kernel 1: per-batch matching + localization loss ------
// One block per batch image. Whole per-batch best-truth tables live in LDS
// (~75 KB) — only possible thanks to CDNA5's 320 KB-per-WGP LDS.
__global__ __launch_bounds__(512) void k_match(
        const float* __restrict__ loc_data,
        const float* __restrict__ priors,
        const float* __restrict__ targets,
        float* __restrict__ accums,
        int*   __restrict__ num_pos,
        int*   __restrict__ conf_t)
{
    __shared__ float              s_bto[NUM_P];      // best truth overlap per prior
    __shared__ int                s_bti[NUM_P];      // best truth index per prior
    __shared__ unsigned long long s_bestp[NUM_O];    // packed (ov_bits<<32)|(~p)
    __shared__ float              s_tr[NUM_O * 4];
    __shared__ int                s_lab[NUM_O];
    __shared__ float              s_redf[512];
    __shared__ int                s_redi[512];

    const int b   = blockIdx.x;
    const int tid = threadIdx.x;
    const int nt  = blockDim.x;

    if (tid < NUM_O) {
        const float* tg = targets + ((size_t)b * NUM_O + tid) * 5;
        s_tr[tid * 4 + 0] = tg[0];
        s_tr[tid * 4 + 1] = tg[1];
        s_tr[tid * 4 + 2] = tg[2];
        s_tr[tid * 4 + 3] = tg[3];
        s_lab[tid]        = (int)tg[4];
        s_bestp[tid]      = 0ULL;
    }
    __syncthreads();

    // Phase 1: IoU of every prior vs every truth; track argmax over truths
    // (first index on ties via strict '>') and per-truth argmax over priors
    // (smallest prior index on ties via inverted-index packing + u64 max).
    for (int p = tid; p < NUM_P; p += nt) {
        float4 pr = ((const float4*)priors)[p];
        float px0 = pr.x - pr.z * 0.5f, py0 = pr.y - pr.w * 0.5f;
        float px1 = pr.x + pr.z * 0.5f, py1 = pr.y + pr.w * 0.5f;
        float areaB = (px1 - px0) * (py1 - py0);
        float best = -1.0f;
        int   bt   = 0;
        for (int t = 0; t < NUM_O; ++t) {
            float tx0 = s_tr[t * 4 + 0], ty0 = s_tr[t * 4 + 1];
            float tx1 = s_tr[t * 4 + 2], ty1 = s_tr[t * 4 + 3];
            float w = fmaxf(fminf(tx1, px1) - fmaxf(tx0, px0), 0.0f);
            float h = fmaxf(fminf(ty1, py1) - fmaxf(ty0, py0), 0.0f);
            float inter = w * h;
            float areaA = (tx1 - tx0) * (ty1 - ty0);
            float ov = inter / (areaA + areaB - inter);
            if (ov > best) { best = ov; bt = t; }
            unsigned long long key =
                ((unsigned long long)__float_as_uint(ov) << 32) |
                (unsigned long long)(0xFFFFFFFFu - (unsigned)p);
            if (key > s_bestp[t]) atomicMax(&s_bestp[t], key);   // ds_max_u64
        }
        s_bto[p] = best;
        s_bti[p] = bt;
    }
    __syncthreads();

    // Forced matches: sequential scatter like the reference .at[].set chain.
    if (tid == 0) {
        for (int t = 0; t < NUM_O; ++t) {
            unsigned pbest = 0xFFFFFFFFu - (unsigned)(s_bestp[t] & 0xFFFFFFFFull);
            s_bto[pbest] = 2.0f;
            s_bti[pbest] = t;
        }
    }
    __syncthreads();

    // Phase 2: conf_t, encoded loc targets, smooth-L1 over positives.
    float lloss = 0.0f;
    int   lpos  = 0;
    for (int p = tid; p < NUM_P; p += nt) {
        float ov = s_bto[p];
        int   t  = s_bti[p];
        int   cf = (ov < 0.5f) ? 0 : s_lab[t];
        conf_t[(size_t)b * NUM_P + p] = cf;
        if (cf > 0) {
            float4 pr  = ((const float4*)priors)[p];
            float tx0 = s_tr[t * 4 + 0], ty0 = s_tr[t * 4 + 1];
            float tx1 = s_tr[t * 4 + 2], ty1 = s_tr[t * 4 + 3];
            float gx = ((tx0 + tx1) * 0.5f - pr.x) / (0.1f * pr.z);
            float gy = ((ty0 + ty1) * 0.5f - pr.y) / (0.1f * pr.w);
            float gw = logf((tx1 - tx0) / pr.z) / 0.2f;
            float gh = logf((ty1 - ty0) / pr.w) / 0.2f;
            float4 ld = ((const float4*)loc_data)[(size_t)b * NUM_P + p];
            float d, s = 0.0f;
            d = fabsf(ld.x - gx); s += (d < 1.0f) ? 0.5f * d * d : d - 0.5f;
            d = fabsf(ld.y - gy); s += (d < 1.0f) ? 0.5f * d * d : d - 0.5f;
            d = fabsf(ld.z - gw); s += (d < 1.0f) ? 0.5f * d * d : d - 0.5f;
            d = fabsf(ld.w - gh); s += (d < 1.0f) ? 0.5f * d * d : d - 0.5f;
            lloss += s;
            lpos  += 1;
        }
    }
    s_redf[tid] = lloss;
    s_redi[tid] = lpos;
    __syncthreads();
    for (int off = nt >> 1; off > 0; off >>= 1) {
        if (tid < off) {
            s_redf[tid] += s_redf[tid + off];
            s_redi[tid] += s_redi[tid + off];
        }
        __syncthreads();
    }
    if (tid == 0) {
        atomicAdd(&accums[0], s_redf[0]);
        num_pos[b] = s_redi[0];
    }
}

// ---------------- Kernel 2: cross-entropy (the 181 MB streaming pass) ---
// One wave32 per (b,p) row of 81 logits: lane-split classes {l, l+32, l+64},
// non-temporal (single-use stream) coalesced loads, wave-wide shfl_xor
// reductions for max / exp-sum / picked logit.
__global__ __launch_bounds__(256) void k_ce(
        const float* __restrict__ conf,
        const int*   __restrict__ conf_t,
        float* __restrict__ accums,
        float* __restrict__ mine)
{
    const long long R    = (long long)NUM_B * NUM_P;
    const int       wid  = threadIdx.x >> 5;
    const int       lane = threadIdx.x & 31;
    const long long row  = (long long)blockIdx.x * 8 + wid;
    if (row >= R) return;

    const float* base = conf + row * NUM_C;
    __builtin_prefetch(base + 8 * NUM_C, 0, 0);   // global_prefetch_b8

    const int c0 = lane, c1 = lane + 32, c2 = lane + 64;
    float v0 = __builtin_nontemporal_load(base + c0);
    float v1 = __builtin_nontemporal_load(base + c1);
    float v2 = (c2 < NUM_C) ? __builtin_nontemporal_load(base + c2) : -3.4e38f;

    float m = fmaxf(v0, fmaxf(v1, v2));
    for (int off = 16; off > 0; off >>= 1) m = fmaxf(m, __shfl_xor(m, off, 32));

    float s = expf(v0 - m) + expf(v1 - m) + ((c2 < NUM_C) ? expf(v2 - m) : 0.0f);
    const int k = conf_t[row];
    float pick = ((c0 == k) ? v0 : 0.0f) + ((c1 == k) ? v1 : 0.0f) +
                 ((c2 < NUM_C && c2 == k) ? v2 : 0.0f);
    for (int off = 16; off > 0; off >>= 1) {
        s    += __shfl_xor(s, off, 32);
        pick += __shfl_xor(pick, off, 32);
    }

    if (lane == 0) {
        float ce = (m + logf(s)) - pick;
        if (k > 0) {                       // positive: counts toward loss_c
            mine[row] = 0.0f;
            atomicAdd(&accums[1], ce);
        } else {
            mine[row] = ce;                // candidate for hard-neg mining
        }
    }
}

// ---------------- Kernel 3: hard-negative mining via exact radix-select -
// Sum of top-k is tie-invariant, so find the k-th largest float bit-pattern
// with 4 byte-wise histogram passes, then sum (> threshold) + r * threshold.
// The per-batch value array is staged into LDS with CDNA5 async loads.
__global__ __launch_bounds__(256) void k_neg(
        const float* __restrict__ mine,
        const int*   __restrict__ num_pos,
        float* __restrict__ accums)
{
    __shared__ float    s_vals[NUM_P];
    __shared__ unsigned s_hist[256];
    __shared__ float    s_red[256];
    __shared__ unsigned s_pref_sh;
    __shared__ unsigned s_rem_sh;

    const int    b   = blockIdx.x;
    const int    tid = threadIdx.x;
    const float* src = mine + (size_t)b * NUM_P;
    const int    P4  = NUM_P / 4;   // 2183 exact

#if HAVE_ASYNC_LDS
    for (int i = tid; i < P4; i += 256) {
        __builtin_amdgcn_global_load_async_to_lds_b128(
            (gptr_v4i)(uintptr_t)(src + 4 * i),
            (lptr_v4i)(uintptr_t)(&s_vals[4 * i]),
            0, 0);
    }
    wait_asynccnt0();
    __syncthreads();
#else
    for (int i = tid; i < P4; i += 256) {
        __builtin_prefetch(src + 4 * i + 1024, 0, 0);
        ((float4*)s_vals)[i] = ((const float4*)src)[i];
    }
    __syncthreads();
#endif

    const int       np  = num_pos[b];
    const long long k64 = 3LL * np;
    const int       k   = (k64 > NUM_P - 1) ? (NUM_P - 1) : (int)k64;
    if (k <= 0) return;

    if (tid == 0) { s_pref_sh = 0u; s_rem_sh = (unsigned)k; }
    __syncthreads();

    for (int pass = 0; pass < 4; ++pass) {
        const int shift = 24 - 8 * pass;
        s_hist[tid] = 0u;
        __syncthreads();
        const unsigned pref   = s_pref_sh;
        const unsigned maskhi = (pass == 0) ? 0u : (0xFFFFFFFFu << (shift + 8));
        for (int j = tid; j < NUM_P; j += 256) {
            unsigned kv = __float_as_uint(s_vals[j]);   // values >= 0: monotone
            if ((kv & maskhi) == (pref & maskhi))
                atomicAdd(&s_hist[(kv >> shift) & 255u], 1u);
        }
        __syncthreads();
        if (tid == 0) {
            unsigned rem = s_rem_sh, cum = 0;
            for (int d = 255; d >= 0; --d) {
                unsigned h = s_hist[d];
                if (cum + h >= rem) {
                    s_rem_sh  = rem - cum;                       // in [1, h]
                    s_pref_sh = pref | ((unsigned)d << shift);
                    break;
                }
                cum += h;
            }
        }
        __syncthreads();
    }

    const unsigned V = s_pref_sh;   // exact bit pattern of k-th largest value
    const unsigned r = s_rem_sh;    // how many copies of V are selected
    float local = 0.0f;
    for (int j = tid; j < NUM_P; j += 256) {
        unsigned kv = __float_as_uint(s_vals[j]);
        if (kv > V) local += s_vals[j];
    }
    s_red[tid] = local;
    __syncthreads();
    for (int off = 128; off > 0; off >>= 1) {
        if (tid < off) s_red[tid] += s_red[tid + off];
        __syncthreads();
    }
    if (tid == 0)
        atomicAdd(&accums[1], s_red[0] + (float)r * __uint_as_float(V));
}

// ---------------- Kernel 4: finalize -----------------------------------
__global__ void k_fin(const float* __restrict__ accums,
                      const int*   __restrict__ num_pos,
                      float* __restrict__ out)
{
    if (threadIdx.x == 0 && blockIdx.x == 0) {
        int N = 0;
        for (int b = 0; b < NUM_B; ++b) N += num_pos[b];
        float n = (float)N;
        out[0] = accums[0] / n;
        out[1] = accums[1] / n;
    }
}

extern "C" void kernel_launch(void* const* d_in, const int* in_sizes, int n_in,
                              void* d_out, int out_size, void* d_ws, size_t ws_size,
                              hipStream_t stream)
{
    const float* loc_data  = (const float*)d_in[0];   // (64, 8732, 4)
    const float* conf_data = (const float*)d_in[1];   // (64, 8732, 81)
    const float* priors    = (const float*)d_in[2];   // (8732, 4)
    const float* targets   = (const float*)d_in[3];   // (64, 50, 5)
    float* out = (float*)d_out;                       // 2 scalars

    char*  ws      = (char*)d_ws;
    float* accums  = (float*)ws;                                   // 2 floats (pad 64B)
    int*   num_pos = (int*)(ws + 64);                              // 64 ints (pad 256B)
    int*   conf_t  = (int*)(ws + 64 + 256);                        // B*P ints
    float* mine    = (float*)(ws + 64 + 256 +
                              sizeof(int) * (size_t)NUM_B * NUM_P);// B*P floats

    k_init<<<1, 64, 0, stream>>>(accums);
    k_match<<<NUM_B, 512, 0, stream>>>(loc_data, priors, targets,
                                       accums, num_pos, conf_t);
    const long long R = (long long)NUM_B * NUM_P;     // 558848, divisible by 8
    k_ce<<<(int)(R / 8), 256, 0, stream>>>(conf_data, conf_t, accums, mine);
    k_neg<<<NUM_B, 256, 0, stream>>>(mine, num_pos, accums);
    k_fin<<<1, 1, 0, stream>>>(accums, num_pos, out);
}